// SNNLoss_43628277793496
// MI455X (gfx1250) — compile-verified
//
#include <hip/hip_runtime.h>
#include <hip/hip_bf16.h>

// SNN loss, B=4096, D=128, T=100, f32.
// Gram matrix via V_WMMA_F32_16X16X4_F32 (CDNA5 f32 matrix op).

typedef float v2f __attribute__((ext_vector_type(2)));
typedef float v8f __attribute__((ext_vector_type(8)));

#define BATCH 4096
#define DIM   128
#define NTILE (BATCH / 16)      // 256 column tiles
#define KSTEPS (DIM / 4)        // 32 wmma k-steps
__constant__ const float kInvT = 1.0f / 100.0f;

// ---------------------------------------------------------------------------
// Kernel 0: per-row squared norms  sq[i] = ||x_i||^2
// ---------------------------------------------------------------------------
__global__ __launch_bounds__(256) void snn_sq_kernel(const float* __restrict__ x,
                                                     float* __restrict__ sq) {
    int r = blockIdx.x * blockDim.x + threadIdx.x;
    if (r < BATCH) {
        const float4* xr = (const float4*)(x + (size_t)r * DIM);
        float s = 0.0f;
#pragma unroll
        for (int k = 0; k < DIM / 4; ++k) {
            float4 v = xr[k];
            s += v.x * v.x + v.y * v.y + v.z * v.z + v.w * v.w;
        }
        sq[r] = s;
    }
}

// ---------------------------------------------------------------------------
// Kernel 1: one block per 16-row strip; 8 waves split the 256 column tiles.
// Each wave computes 16x16 Gram tiles with f32 WMMA, turns them into
// exp(-dist/T), and accumulates per-row num/den. LDS combine -> row loss.
// ---------------------------------------------------------------------------
__global__ __launch_bounds__(256) void snn_row_kernel(const float* __restrict__ x,
                                                      const int*   __restrict__ y,
                                                      const float* __restrict__ sq,
                                                      float*       __restrict__ rowloss) {
    __shared__ float lds_num[8 * 16];
    __shared__ float lds_den[8 * 16];

    const int tid  = threadIdx.x;
    const int wave = tid >> 5;
    const int lane = tid & 31;
    const int half = lane >> 4;   // 0: M 0..7 / K 0..1   1: M 8..15 / K 2..3
    const int lq   = lane & 15;
    const int row0 = blockIdx.x * 16;

    // --- Hoist A fragments: depend only on the row strip -------------------
    // ISA 16x4 f32 A layout: lanes 0-15 hold (K=k0, K=k0+1) for M=lane;
    // lanes 16-31 hold (K=k0+2, K=k0+3).
    v2f a[KSTEPS];
    {
        const float* arow = x + (size_t)(row0 + lq) * DIM + 2 * half;
#pragma unroll
        for (int kk = 0; kk < KSTEPS; ++kk) {
            a[kk].x = arow[kk * 4 + 0];
            a[kk].y = arow[kk * 4 + 1];
        }
    }

    // Row metadata for the 8 M-values this lane's C fragment covers.
    float sqi[8];
    int   yi[8];
#pragma unroll
    for (int v = 0; v < 8; ++v) {
        const int m = v + 8 * half;
        sqi[v] = sq[row0 + m];
        yi[v]  = y[row0 + m];
    }

    float num_acc[8], den_acc[8];
#pragma unroll
    for (int v = 0; v < 8; ++v) { num_acc[v] = 0.0f; den_acc[v] = 0.0f; }

    // --- Column-tile loop ---------------------------------------------------
    for (int ct = wave; ct < NTILE; ct += 8) {
        const int col0 = ct * 16;
        const float* brow = x + (size_t)(col0 + lq) * DIM + 2 * half;

        // Prefetch our next B tile (same lane pattern, 8 tiles ahead).
        if (ct + 8 < NTILE)
            __builtin_prefetch(x + (size_t)(col0 + 128 + lq) * DIM, 0, 1);

        // B tile of x^T loads with exactly the A-fragment lane pattern:
        // B[k][n] = x[col0+n][k] -> VGPR0: K=2*half, VGPR1: K=2*half+1, N=lq.
        v8f c0 = {};
        v8f c1 = {};
#pragma unroll
        for (int kk = 0; kk < KSTEPS; kk += 2) {
            v2f b0; b0.x = brow[kk * 4 + 0]; b0.y = brow[kk * 4 + 1];
            v2f b1; b1.x = brow[kk * 4 + 4]; b1.y = brow[kk * 4 + 5];
            c0 = __builtin_amdgcn_wmma_f32_16x16x4_f32(false, a[kk],     false, b0,
                                                       (short)0, c0, false, false);
            c1 = __builtin_amdgcn_wmma_f32_16x16x4_f32(false, a[kk + 1], false, b1,
                                                       (short)0, c1, false, false);
        }

        const float sqj = sq[col0 + lq];
        const int   yj  = y[col0 + lq];
        const int   j   = col0 + lq;

#pragma unroll
        for (int v = 0; v < 8; ++v) {
            const int i  = row0 + v + 8 * half;
            float g      = c0[v] + c1[v];
            float dist   = sqi[v] + sqj - 2.0f * g;
            dist         = fmaxf(dist, 0.0f);          // clamp fp error; diag -> 0
            const float e = __expf(-dist * kInvT);
            den_acc[v] += e;                           // all j, incl. j==i (e=1)
            if (yi[v] == yj && i != j) num_acc[v] += e;
        }
    }

    // --- Reduce across N (the 16 lanes of each half) ------------------------
#pragma unroll
    for (int off = 1; off < 16; off <<= 1) {
#pragma unroll
        for (int v = 0; v < 8; ++v) {
            num_acc[v] += __shfl_xor(num_acc[v], off, 32);
            den_acc[v] += __shfl_xor(den_acc[v], off, 32);
        }
    }
    if (lq == 0) {
#pragma unroll
        for (int v = 0; v < 8; ++v) {
            const int m = v + 8 * half;
            lds_num[wave * 16 + m] = num_acc[v];
            lds_den[wave * 16 + m] = den_acc[v];
        }
    }
    __syncthreads();

    // --- Combine the 8 waves; one row loss per thread 0..15 -----------------
    if (tid < 16) {
        float n = 0.0f, d = 0.0f;
#pragma unroll
        for (int w = 0; w < 8; ++w) {
            n += lds_num[w * 16 + tid];
            d += lds_den[w * 16 + tid];
        }
        rowloss[row0 + tid] = __logf(d) - __logf(n);   // -log(num/den)
    }
}

// ---------------------------------------------------------------------------
// Kernel 2: deterministic mean over the 4096 row losses.
// ---------------------------------------------------------------------------
__global__ __launch_bounds__(1024) void snn_reduce_kernel(const float* __restrict__ rowloss,
                                                          float* __restrict__ out) {
    __shared__ float s[1024];
    float acc = 0.0f;
    for (int i = threadIdx.x; i < BATCH; i += 1024) acc += rowloss[i];
    s[threadIdx.x] = acc;
    __syncthreads();
#pragma unroll
    for (int off = 512; off > 0; off >>= 1) {
        if (threadIdx.x < off) s[threadIdx.x] += s[threadIdx.x + off];
        __syncthreads();
    }
    if (threadIdx.x == 0) out[0] = s[0] * (1.0f / (float)BATCH);
}

// ---------------------------------------------------------------------------
extern "C" void kernel_launch(void* const* d_in, const int* in_sizes, int n_in,
                              void* d_out, int out_size, void* d_ws, size_t ws_size,
                              hipStream_t stream) {
    (void)in_sizes; (void)n_in; (void)out_size; (void)ws_size;

    const float* x = (const float*)d_in[0];
    const int*   y = (const int*)d_in[1];
    float*       out = (float*)d_out;

    float* sq      = (float*)d_ws;            // 4096 floats
    float* rowloss = sq + BATCH;              // 4096 floats  (32 KB total)

    snn_sq_kernel<<<BATCH / 256, 256, 0, stream>>>(x, sq);
    snn_row_kernel<<<BATCH / 16, 256, 0, stream>>>(x, y, sq, rowloss);
    snn_reduce_kernel<<<1, 1024, 0, stream>>>(rowloss, out);
}